// STGCN_block_240518168586
// MI455X (gfx1250) — compile-verified
//
#include <hip/hip_runtime.h>
#include <math.h>

typedef __attribute__((ext_vector_type(2))) float v2f;
typedef __attribute__((ext_vector_type(8))) float v8f;

#define Bn 2048
#define Tn 3
#define Vn 62
#define Fn 5
#define Hn 64
#define VP 64
#define FP 8
#define NTHREADS 256

#define OFF_SLOSS ((size_t)8126464)   // B*V*H2
#define OFF_DLOSS ((size_t)8126465)
#define OFF_S     ((size_t)8126466)

#define ALPHA_C     0.0001f
#define LN_EPS_C    1e-5f
#define INV_SQRT310 0.05679618342470648f
#define INV_SQRT15  0.25819888974716115f

__global__ __launch_bounds__(NTHREADS, 1)
void stgcn_main_kernel(const float* __restrict__ x,   const float* __restrict__ U1,
                       const float* __restrict__ U2,  const float* __restrict__ U3,
                       const float* __restrict__ be,  const float* __restrict__ Ve,
                       const float* __restrict__ W1,  const float* __restrict__ W2,
                       const float* __restrict__ W3,  const float* __restrict__ bs,
                       const float* __restrict__ Vs,  const float* __restrict__ aP,
                       const float* __restrict__ Theta, const float* __restrict__ tw,
                       const float* __restrict__ tb,  const float* __restrict__ rw,
                       const float* __restrict__ rb,  const float* __restrict__ gamma,
                       const float* __restrict__ beta,
                       float* __restrict__ out, float* __restrict__ ws)
{
  __shared__ float xs[Tn][VP][FP];      // x[b], zero-padded
  __shared__ float xT[Tn][VP][FP];      // x_TAt
  __shared__ float Pm[VP][VP];          // sigmoid(s_prod+bs) -> later L_t
  __shared__ float Qm[VP][VP];          // Sm -> spatial_At
  __shared__ float Rm[VP][VP];          // Vs (padded) -> tmpS -> S -> Tk -> z
  __shared__ float sg[Hn*Tn][VP];       // spatial_gcn in B-layout sg[c*3+t][v]
  __shared__ float tmpk[Tn][VP][FP];
  // staged constants
  __shared__ float ThL[3*Fn*Hn];        // Theta
  __shared__ float rwL[Hn*Fn];
  __shared__ float U2L[Fn*Vn];
  __shared__ float tbL[Hn], rbL[Hn], gmL[Hn], btL[Hn];
  __shared__ float U1L[VP], U3L[FP], W1L[4], W2L[16], W3L[FP], aL[FP], VeL[12], beL[12];
  // small intermediates
  __shared__ float xU1[Tn][FP];
  __shared__ float lhsS[Tn][VP];
  __shared__ float rhsS[VP][4];
  __shared__ float sigS[4][4];
  __shared__ float tAtS[Tn][4];
  __shared__ float slhsS[VP][4];
  __shared__ float srhsS[Tn][VP];
  __shared__ float csum[VP];
  __shared__ float DgS[VP];
  __shared__ float red[NTHREADS];

  const int tid = threadIdx.x;
  const int b   = blockIdx.x;
  const float* xb = x + (size_t)b * (Tn * Vn * Fn);

  // ---------------- Stage 0: stage x, Vs (zero padded) and small constants ----------------
  for (int idx = tid; idx < Tn * VP * FP; idx += NTHREADS) {
    int t = idx / (VP * FP), r = idx % (VP * FP), v = r / FP, f = r % FP;
    xs[t][v][f] = (v < Vn && f < Fn) ? xb[(t * Vn + v) * Fn + f] : 0.f;
  }
  for (int idx = tid; idx < VP * VP; idx += NTHREADS) {
    int m = idx / VP, k = idx % VP;
    Rm[m][k] = (m < Vn && k < Vn) ? Vs[m * Vn + k] : 0.f;   // Rm = padded Vs (A-matrix)
  }
  for (int i = tid; i < 3 * Fn * Hn; i += NTHREADS) ThL[i] = Theta[i];
  for (int i = tid; i < Hn * Fn; i += NTHREADS)     rwL[i] = rw[i];
  for (int i = tid; i < Fn * Vn; i += NTHREADS)     U2L[i] = U2[i];
  if (tid < Hn) { tbL[tid] = tb[tid]; rbL[tid] = rb[tid]; gmL[tid] = gamma[tid]; btL[tid] = beta[tid]; }
  if (tid < Vn) U1L[tid] = U1[tid];
  if (tid < Fn) { U3L[tid] = U3[tid]; W3L[tid] = W3[tid]; aL[tid] = aP[tid]; }
  if (tid < Tn) W1L[tid] = W1[tid];
  if (tid < 15) W2L[tid] = W2[tid];
  if (tid < 9)  { VeL[tid] = Ve[tid]; beL[tid] = be[tid]; }
  __syncthreads();

  // ---------------- Stage 1: temporal attention ----------------
  for (int idx = tid; idx < Tn * Fn; idx += NTHREADS) {
    int t = idx / Fn, f = idx % Fn;
    float s = 0.f;
    for (int v = 0; v < Vn; ++v) s += xs[t][v][f] * U1L[v];
    xU1[t][f] = s;
  }
  __syncthreads();
  for (int idx = tid; idx < Tn * Vn; idx += NTHREADS) {
    int t = idx / Vn, u = idx % Vn;
    float s = 0.f, r = 0.f;
    for (int f = 0; f < Fn; ++f) { s += xU1[t][f] * U2L[f * Vn + u]; r += U3L[f] * xs[t][u][f]; }
    lhsS[t][u] = s;
    rhsS[u][t] = r;
  }
  __syncthreads();
  if (tid < 9) {                      // sigmoid(prod + be)
    int t = tid / 3, u = tid % 3;
    float s = 0.f;
    for (int v = 0; v < Vn; ++v) s += lhsS[t][v] * rhsS[v][u];
    s += beL[t * 3 + u];
    sigS[t][u] = 1.f / (1.f + __expf(-s));
  }
  __syncthreads();
  if (tid < 3) {                      // E = Ve @ sig ; softmax over t (axis=1)
    int u = tid;
    float e[3], m = -1e30f, sum = 0.f;
    for (int t = 0; t < 3; ++t) {
      float s = 0.f;
      for (int k = 0; k < 3; ++k) s += VeL[t * 3 + k] * sigS[k][u];
      e[t] = s; m = fmaxf(m, s);
    }
    for (int t = 0; t < 3; ++t) { e[t] = __expf(e[t] - m); sum += e[t]; }
    float inv = 1.f / sum;
    for (int t = 0; t < 3; ++t) tAtS[t][u] = e[t] * inv;
  }
  __syncthreads();
  for (int idx = tid; idx < Tn * VP * FP; idx += NTHREADS) {   // x_TAt
    int u = idx / (VP * FP), r = idx % (VP * FP), v = r / FP, f = r % FP;
    xT[u][v][f] = (xs[0][v][f] * tAtS[0][u] + xs[1][v][f] * tAtS[1][u] +
                   xs[2][v][f] * tAtS[2][u]) * INV_SQRT310;
  }
  __syncthreads();

  // ---------------- Stage 2: spatial attention inputs ----------------
  for (int v = tid; v < Vn; v += NTHREADS) {
    float y[Fn];
    for (int f = 0; f < Fn; ++f)
      y[f] = xT[0][v][f] * W1L[0] + xT[1][v][f] * W1L[1] + xT[2][v][f] * W1L[2];
    for (int s = 0; s < 3; ++s) {
      float acc = 0.f;
      for (int f = 0; f < Fn; ++f) acc += y[f] * W2L[f * 3 + s];
      slhsS[v][s] = acc;
    }
  }
  for (int idx = tid; idx < Tn * Vn; idx += NTHREADS) {
    int t = idx / Vn, v = idx % Vn;
    float s = 0.f;
    for (int f = 0; f < Fn; ++f) s += W3L[f] * xT[t][v][f];
    srhsS[t][v] = s;
  }
  __syncthreads();
  for (int idx = tid; idx < VP * VP; idx += NTHREADS) {   // sigmoid(s_prod+bs), zero pad
    int u = idx / VP, v = idx % VP;
    float val = 0.f;
    if (u < Vn && v < Vn) {
      float s = slhsS[u][0] * srhsS[0][v] + slhsS[u][1] * srhsS[1][v] +
                slhsS[u][2] * srhsS[2][v] + bs[u * Vn + v];
      val = 1.f / (1.f + __expf(-s));
    }
    Pm[u][v] = val;
  }
  __syncthreads();

  // ---------------- Stage 3: Sm = Vs @ Pm via V_WMMA_F32_16X16X4_F32 (pipelined) ----------------
  {
    const int wave = tid >> 5, lane = tid & 31, n0 = lane & 15, half = lane >> 4;
    for (int tile = wave * 2; tile < wave * 2 + 2; ++tile) {
      const int tm = tile >> 2, tn = tile & 3;
      const int m  = tm * 16 + n0;     // A row (u)
      const int nv = tn * 16 + n0;     // B/D col (v)
      v8f c = {};
      v2f a = *(const v2f*)&Rm[m][half * 2];            // Vs staged in Rm, zero padded
      v2f bb; bb.x = Pm[half][nv]; bb.y = Pm[half + 2][nv];
#pragma unroll
      for (int ks = 0; ks < 16; ++ks) {
        v2f an = a, bn = bb;
        if (ks + 1 < 16) {
          const int ka = (ks + 1) * 4 + half * 2;
          const int kb = (ks + 1) * 4 + half;
          an = *(const v2f*)&Rm[m][ka];
          bn.x = Pm[kb][nv]; bn.y = Pm[kb + 2][nv];
        }
        c = __builtin_amdgcn_wmma_f32_16x16x4_f32(false, a, false, bb,
                                                  (short)0, c, false, false);
        a = an; bb = bn;
      }
      for (int r = 0; r < 8; ++r)
        Qm[tm * 16 + r + half * 8][nv] = c[r];
    }
  }
  __syncthreads();
  for (int v = tid; v < Vn; v += NTHREADS) {    // softmax over u (axis=1)
    float m = -1e30f;
    for (int u = 0; u < Vn; ++u) m = fmaxf(m, Qm[u][v]);
    float s = 0.f;
    for (int u = 0; u < Vn; ++u) { float e = __expf(Qm[u][v] - m); Qm[u][v] = e; s += e; }
    float inv = 1.f / s;
    for (int u = 0; u < Vn; ++u) Qm[u][v] *= inv;
  }
  __syncthreads();

  // ---------------- Stage 4: adaptive graph S, losses, L_t ----------------
  for (int idx = tid; idx < VP * VP; idx += NTHREADS) ((float*)Rm)[idx] = 0.f;
  __syncthreads();
  float dsum = 0.f;
  for (int idx = tid; idx < Vn * Vn; idx += NTHREADS) {
    int i = idx / Vn, j = idx % Vn;
    float e = 0.f, dd = 0.f;
    for (int f = 0; f < Fn; ++f) {
      float d = xs[1][i][f] - xs[1][j][f];
      e  += fabsf(d) * aL[f];
      dd += d * d;
    }
    Rm[i][j] = __expf(fmaxf(e, 0.f));
    dsum += dd;                              // dloss = ALPHA * sum(d2) (cols of S sum to 1)
  }
  __syncthreads();
  for (int j = tid; j < Vn; j += NTHREADS) {
    float s = 0.f;
    for (int i = 0; i < Vn; ++i) s += Rm[i][j];
    csum[j] = 1.f / s;
  }
  __syncthreads();
  float ssum = 0.f;
  for (int idx = tid; idx < Vn * Vn; idx += NTHREADS) {
    int i = idx / Vn, j = idx % Vn;
    float sv = Rm[i][j] * csum[j];
    Rm[i][j] = sv;
    ssum += sv * sv;
    out[OFF_S + ((size_t)b * Vn + i) * Vn + j] = sv;
  }
  __syncthreads();
  for (int j = tid; j < Vn; j += NTHREADS) {
    float s = 0.f;
    for (int i = 0; i < Vn; ++i) s += Rm[i][j];
    DgS[j] = s;
  }
  __syncthreads();
  for (int idx = tid; idx < VP * VP; idx += NTHREADS) {   // L_t -> Pm (zero pad)
    int i = idx / VP, j = idx % VP;
    float val = 0.f;
    if (i < Vn && j < Vn) val = ((i == j) ? (DgS[j] - 1.f) : 0.f) - Rm[i][j];
    Pm[i][j] = val;
  }
  red[tid] = dsum; __syncthreads();
  for (int s = NTHREADS / 2; s > 0; s >>= 1) { if (tid < s) red[tid] += red[tid + s]; __syncthreads(); }
  if (tid == 0) ws[2 * (size_t)b + 1] = red[0];
  __syncthreads();
  red[tid] = ssum; __syncthreads();
  for (int s = NTHREADS / 2; s > 0; s >>= 1) { if (tid < s) red[tid] += red[tid + s]; __syncthreads(); }
  if (tid == 0) ws[2 * (size_t)b] = red[0];
  __syncthreads();

  // ---------------- Stage 5: Chebyshev GCN -> sg[c*3+t][v] ----------------
  for (int idx = tid; idx < Hn * Tn * VP; idx += NTHREADS) ((float*)sg)[idx] = 0.f;
  __syncthreads();
  for (int k = 0; k < 3; ++k) {
    for (int idx = tid; idx < VP * VP; idx += NTHREADS) {   // Tk = cheb_k * At -> Rm
      int u = idx / VP, v = idx % VP;
      float lt  = Pm[u][v];
      float eye = (u == v) ? 1.f : 0.f;
      float ch  = (k == 0) ? eye : ((k == 1) ? lt : (2.f * lt * lt - eye));
      Rm[u][v] = ch * Qm[u][v];
    }
    __syncthreads();
    for (int idx = tid; idx < Tn * Vn * Fn; idx += NTHREADS) {  // tmp[t][v][f]
      int t = idx / (Vn * Fn), r = idx % (Vn * Fn), v = r / Fn, f = r % Fn;
      float s = 0.f;
      for (int u = 0; u < Vn; ++u) s += Rm[u][v] * xs[t][u][f];
      tmpk[t][v][f] = s;
    }
    __syncthreads();
    for (int idx = tid; idx < Hn * Tn * Vn; idx += NTHREADS) {  // += tmp @ Theta[k]
      int o = idx / (Tn * Vn), r = idx % (Tn * Vn), t = r / Vn, v = r % Vn;
      float s = 0.f;
      for (int f = 0; f < Fn; ++f) s += tmpk[t][v][f] * ThL[(k * Fn + f) * Hn + o];
      sg[o * Tn + t][v] += s;
    }
    __syncthreads();
  }
  for (int idx = tid; idx < Hn * Tn * VP; idx += NTHREADS) {    // relu
    float z = ((float*)sg)[idx];
    ((float*)sg)[idx] = fmaxf(z, 0.f);
  }
  __syncthreads();

  // ---------------- Stage 6: tc = tw(64x192) @ sg(192x62) via WMMA (pipelined) ----------------
  {
    const int wave = tid >> 5, lane = tid & 31, n0 = lane & 15, half = lane >> 4;
    for (int tile = wave * 2; tile < wave * 2 + 2; ++tile) {
      const int tm = tile >> 2, tn = tile & 3;
      const int m  = tm * 16 + n0;     // o row
      const int nv = tn * 16 + n0;     // v col
      const float* twRow = tw + (size_t)m * 192;
      v8f c = {};
      v2f a = *(const v2f*)(twRow + half * 2);
      v2f bb; bb.x = sg[half][nv]; bb.y = sg[half + 2][nv];
#pragma unroll 4
      for (int ks = 0; ks < 48; ++ks) {
        v2f an = a, bn = bb;
        if (ks + 1 < 48) {
          const int ka = (ks + 1) * 4 + half * 2;
          const int kb = (ks + 1) * 4 + half;
          an = *(const v2f*)(twRow + ka);
          bn.x = sg[kb][nv]; bn.y = sg[kb + 2][nv];
        }
        c = __builtin_amdgcn_wmma_f32_16x16x4_f32(false, a, false, bb,
                                                  (short)0, c, false, false);
        a = an; bb = bn;
      }
      for (int r = 0; r < 8; ++r) {    // z = relu(res + tc) -> Rm (reused as z buffer)
        int o = tm * 16 + r + half * 8;
        float tcv = (c[r] + tbL[o]) * INV_SQRT15;
        float res = rbL[o];
        for (int f = 0; f < Fn; ++f) res += xs[0][nv][f] * rwL[o * Fn + f];
        Rm[o][nv] = fmaxf(res + tcv, 0.f);
      }
    }
  }
  __syncthreads();

  // ---------------- Stage 7: LayerNorm over H2 per (b, v) ----------------
  for (int v = tid; v < Vn; v += NTHREADS) {
    float mu = 0.f;
    for (int o = 0; o < Hn; ++o) mu += Rm[o][v];
    mu *= (1.f / Hn);
    float var = 0.f;
    for (int o = 0; o < Hn; ++o) { float d = Rm[o][v] - mu; var += d * d; }
    var *= (1.f / Hn);
    float inv = rsqrtf(var + LN_EPS_C);
    size_t base = ((size_t)b * Vn + v) * Hn;
    for (int o = 0; o < Hn; ++o)
      out[base + o] = (Rm[o][v] - mu) * inv * gmL[o] + btL[o];
  }
}

__global__ __launch_bounds__(256, 1)
void stgcn_reduce_kernel(const float* __restrict__ ws, float* __restrict__ out)
{
  __shared__ float r0[256], r1[256];
  int tid = threadIdx.x;
  float s0 = 0.f, s1 = 0.f;
  for (int i = tid; i < Bn; i += 256) { s0 += ws[2 * i]; s1 += ws[2 * i + 1]; }
  r0[tid] = s0; r1[tid] = s1;
  __syncthreads();
  for (int s = 128; s > 0; s >>= 1) {
    if (tid < s) { r0[tid] += r0[tid + s]; r1[tid] += r1[tid + s]; }
    __syncthreads();
  }
  if (tid == 0) {
    out[OFF_SLOSS] = (ALPHA_C / (float)Bn) * r0[0];
    out[OFF_DLOSS] = ALPHA_C * r1[0];
  }
}

extern "C" void kernel_launch(void* const* d_in, const int* in_sizes, int n_in,
                              void* d_out, int out_size, void* d_ws, size_t ws_size,
                              hipStream_t stream) {
  (void)in_sizes; (void)n_in; (void)out_size; (void)ws_size;
  const float* x     = (const float*)d_in[0];
  const float* U1    = (const float*)d_in[1];
  const float* U2    = (const float*)d_in[2];
  const float* U3    = (const float*)d_in[3];
  const float* be    = (const float*)d_in[4];
  const float* Ve    = (const float*)d_in[5];
  const float* W1    = (const float*)d_in[6];
  const float* W2    = (const float*)d_in[7];
  const float* W3    = (const float*)d_in[8];
  const float* bs    = (const float*)d_in[9];
  const float* Vs    = (const float*)d_in[10];
  const float* aP    = (const float*)d_in[11];
  const float* Theta = (const float*)d_in[12];
  const float* tw    = (const float*)d_in[13];
  const float* tb    = (const float*)d_in[14];
  const float* rw    = (const float*)d_in[15];
  const float* rb    = (const float*)d_in[16];
  const float* gamma = (const float*)d_in[17];
  const float* beta  = (const float*)d_in[18];
  float* out = (float*)d_out;
  float* ws  = (float*)d_ws;

  stgcn_main_kernel<<<Bn, NTHREADS, 0, stream>>>(x, U1, U2, U3, be, Ve, W1, W2, W3,
                                                 bs, Vs, aP, Theta, tw, tb, rw, rb,
                                                 gamma, beta, out, ws);
  stgcn_reduce_kernel<<<1, 256, 0, stream>>>(ws, out);
}